// SpecBranch_89601607729227
// MI455X (gfx1250) — compile-verified
//
#include <hip/hip_runtime.h>
#include <cstdint>
#include <cstddef>

// ---------------------------------------------------------------------------
// ChebConv (K=3) two-layer GNN for MI455X (gfx1250, wave32, WMMA).
//   layer1: h   = relu( sum_k Txk @ W1[k] + b1 ),  Tx on 96 features
//   layer2: out =       sum_k Tyk @ W2[k] + b2,    Ty on 64 features
// Sparse propagation is L2-resident scatter-add; dense maps use
// V_WMMA_F32_16X16X4_F32 (fp32 matrix pipe, matches reference precision).
// ---------------------------------------------------------------------------

#define NNODES 50000
#define NEDGES 800000
#define DIN    96
#define DHID   64
#define DOUT   16

typedef float v2f __attribute__((ext_vector_type(2)));
typedef float v8f __attribute__((ext_vector_type(8)));

// ---------------- small utility kernels ----------------
__global__ void zero_f32(float* __restrict__ p, int n) {
  int i = blockIdx.x * blockDim.x + threadIdx.x;
  if (i < n) p[i] = 0.0f;
}

__global__ void neg_copy(float* __restrict__ dst, const float* __restrict__ src, int n) {
  int i = blockIdx.x * blockDim.x + threadIdx.x;
  if (i < n) dst[i] = -src[i];
}

// deg[src] += 1 over all edges (segment_sum of ones over src)
__global__ void edge_degree(const long long* __restrict__ ei, float* __restrict__ deg, int E) {
  int e = blockIdx.x * blockDim.x + threadIdx.x;
  if (e < E) atomicAdd(&deg[(int)ei[e]], 1.0f);
}

// dis[i] = deg>0 ? rsqrt(deg) : 0
__global__ void node_dis(const float* __restrict__ deg, float* __restrict__ dis, int n) {
  int i = blockIdx.x * blockDim.x + threadIdx.x;
  if (i < n) {
    float dg = deg[i];
    dis[i] = (dg > 0.0f) ? rsqrtf(dg) : 0.0f;
  }
}

// w[e] = -dis[src]*dis[dst]
__global__ void edge_norm(const long long* __restrict__ ei, const float* __restrict__ dis,
                          float* __restrict__ w, int E) {
  int e = blockIdx.x * blockDim.x + threadIdx.x;
  if (e < E) {
    int s = (int)ei[e];
    int d = (int)ei[(size_t)E + e];
    w[e] = -dis[s] * dis[d];
  }
}

// out[dst] += scale * w[e] * h[src]  — one thread per (edge, 4 features).
// Gathers are float4; feature matrices fit in 192MB L2, so these and the
// f32 atomics stay on-chip.
template <int D>
__global__ void cheb_prop(const float* __restrict__ h, const long long* __restrict__ ei,
                          const float* __restrict__ w, float* __restrict__ out,
                          float scale, int E) {
  constexpr int Q = D / 4;
  long long tid = (long long)blockIdx.x * blockDim.x + threadIdx.x;
  int e = (int)(tid / Q);
  int q = (int)(tid % Q);
  if (e >= E) return;
  int s = (int)ei[e];
  int d = (int)ei[(size_t)E + e];
  float ww = scale * w[e];
  const float4 v = *(const float4*)(h + (size_t)s * D + q * 4);
  float* o = out + (size_t)d * D + q * 4;
  atomicAdd(o + 0, ww * v.x);
  atomicAdd(o + 1, ww * v.y);
  atomicAdd(o + 2, ww * v.z);
  atomicAdd(o + 3, ww * v.w);
}

// Fused Chebyshev GEMM: out = T0@W[0] + T1@W[1] + T2@W[2] + bias [, relu]
// One wave per 16-row M-tile, covering all NOUT columns (NOUT/16 accumulators)
// so each A fragment feeds NOUT/16 WMMAs. M must be a multiple of 16 (50000 is).
// Fragment layouts per CDNA5 ISA §7.12.2 (f32 16x16x4).
template <int KD, int NOUT>
__global__ void cheb_gemm(const float* __restrict__ T0, const float* __restrict__ T1,
                          const float* __restrict__ T2, const float* __restrict__ W,
                          const float* __restrict__ bias, float* __restrict__ out,
                          int M, int do_relu) {
  constexpr int NT = NOUT / 16;
  int wave   = blockIdx.x * (blockDim.x >> 5) + (threadIdx.x >> 5);
  int mtiles = M >> 4;
  if (wave >= mtiles) return;           // wave-uniform, EXEC stays all-1s

  int lane = threadIdx.x & 31;
  int lm   = lane & 15;   // column / row-in-tile selector
  int lh   = lane >> 4;   // half-wave selector
  int m    = (wave << 4) + lm;

  const float* Ts[3] = {T0, T1, T2};

  v8f acc[NT];
  v8f zero = {0.f, 0.f, 0.f, 0.f, 0.f, 0.f, 0.f, 0.f};
#pragma unroll
  for (int nt = 0; nt < NT; ++nt) acc[nt] = zero;

  for (int t = 0; t < 3; ++t) {
    const float* __restrict__ T  = Ts[t];
    const float* __restrict__ Wt = W + (size_t)t * KD * NOUT;
    for (int k0 = 0; k0 < KD; k0 += 4) {
      int k = k0 + 2 * lh;  // lanes 0-15 hold K={k0,k0+1}; lanes 16-31 K={k0+2,k0+3}
      float2 av = *(const float2*)(T + (size_t)m * KD + k);
      v2f a;
      a.x = av.x;
      a.y = av.y;
#pragma unroll
      for (int nt = 0; nt < NT; ++nt) {
        int n = nt * 16 + lm;
        v2f b;
        b.x = Wt[(size_t)k * NOUT + n];
        b.y = Wt[(size_t)(k + 1) * NOUT + n];
        acc[nt] = __builtin_amdgcn_wmma_f32_16x16x4_f32(
            /*neg_a=*/false, a, /*neg_b=*/false, b,
            /*c_mod=*/(short)0, acc[nt], /*reuse_a=*/false, /*reuse_b=*/false);
      }
    }
  }

  // Epilogue: C/D layout — VGPR v holds row (v + 8*lh), column lm within tile.
#pragma unroll
  for (int nt = 0; nt < NT; ++nt) {
    int col  = nt * 16 + lm;
    float bv = bias[col];
#pragma unroll
    for (int v = 0; v < 8; ++v) {
      int row   = (wave << 4) + v + 8 * lh;
      float val = acc[nt][v] + bv;
      if (do_relu) val = val > 0.0f ? val : 0.0f;
      out[(size_t)row * NOUT + col] = val;
    }
  }
}

// ---------------------------------------------------------------------------
extern "C" void kernel_launch(void* const* d_in, const int* in_sizes, int n_in,
                              void* d_out, int out_size, void* d_ws, size_t ws_size,
                              hipStream_t stream) {
  const float*     x  = (const float*)d_in[0];
  const long long* ei = (const long long*)d_in[1];  // int64 [2, E]
  const float*     W1 = (const float*)d_in[2];      // [3, 96, 64]
  const float*     b1 = (const float*)d_in[3];      // [64]
  const float*     W2 = (const float*)d_in[4];      // [3, 64, 16]
  const float*     b2 = (const float*)d_in[5];      // [16]
  float*           out = (float*)d_out;             // [50000, 16]

  // Workspace carve-out (fp32): deg, dis, w, Tx1, Tx2, h  (~55 MB)
  float* ws    = (float*)d_ws;
  float* deg   = ws;  ws += NNODES;
  float* dis   = ws;  ws += NNODES;
  float* wn    = ws;  ws += NEDGES;
  float* Tx1   = ws;  ws += (size_t)NNODES * DIN;
  float* Tx2   = ws;  ws += (size_t)NNODES * DIN;
  float* hbuf  = ws;  ws += (size_t)NNODES * DHID;
  float* Ty1   = Tx1;  // layer-2 scratch overlays layer-1 scratch
  float* Ty2   = Tx2;

  const int B = 256;
  auto g = [](long long n, int b) { return (int)((n + b - 1) / b); };

  // --- graph normalization ---
  zero_f32<<<g(NNODES, B), B, 0, stream>>>(deg, NNODES);
  edge_degree<<<g(NEDGES, B), B, 0, stream>>>(ei, deg, NEDGES);
  node_dis<<<g(NNODES, B), B, 0, stream>>>(deg, dis, NNODES);
  edge_norm<<<g(NEDGES, B), B, 0, stream>>>(ei, dis, wn, NEDGES);

  // --- layer 1: Tx1 = L x ; Tx2 = 2 L Tx1 - x ---
  zero_f32<<<g((long long)NNODES * DIN, B), B, 0, stream>>>(Tx1, NNODES * DIN);
  cheb_prop<DIN><<<g((long long)NEDGES * (DIN / 4), B), B, 0, stream>>>(x, ei, wn, Tx1, 1.0f, NEDGES);
  neg_copy<<<g((long long)NNODES * DIN, B), B, 0, stream>>>(Tx2, x, NNODES * DIN);
  cheb_prop<DIN><<<g((long long)NEDGES * (DIN / 4), B), B, 0, stream>>>(Tx1, ei, wn, Tx2, 2.0f, NEDGES);

  {
    int mtiles = NNODES / 16;              // 3125 exact
    int wpb    = B / 32;                   // 8 waves/block
    cheb_gemm<DIN, DHID><<<g(mtiles, wpb), B, 0, stream>>>(x, Tx1, Tx2, W1, b1, hbuf, NNODES, 1);
  }

  // --- layer 2 ---
  zero_f32<<<g((long long)NNODES * DHID, B), B, 0, stream>>>(Ty1, NNODES * DHID);
  cheb_prop<DHID><<<g((long long)NEDGES * (DHID / 4), B), B, 0, stream>>>(hbuf, ei, wn, Ty1, 1.0f, NEDGES);
  neg_copy<<<g((long long)NNODES * DHID, B), B, 0, stream>>>(Ty2, hbuf, NNODES * DHID);
  cheb_prop<DHID><<<g((long long)NEDGES * (DHID / 4), B), B, 0, stream>>>(Ty1, ei, wn, Ty2, 2.0f, NEDGES);

  {
    int mtiles = NNODES / 16;
    int wpb    = B / 32;
    cheb_gemm<DHID, DOUT><<<g(mtiles, wpb), B, 0, stream>>>(hbuf, Ty1, Ty2, W2, b2, out, NNODES, 0);
  }
}